// AUX_IVA_ISS_ONLINE_47510928229057
// MI455X (gfx1250) — compile-verified
//
#include <hip/hip_runtime.h>

#define KCH 4
#define TT 2000
#define FF 1025
#define NIT 2
#define ALPHA 0.96f
#define EPSR 1e-10f

#define WGSZ 512
#define WAVES_PER_WG 16
#define NPAIR ((FF + 1) / 2)                                  // 513
#define NWG ((NPAIR + WAVES_PER_WG - 1) / WAVES_PER_WG)       // 33
#define PART_BASE 64                                          // float offset of slot 0
#define PART_STRIDE 160                                       // floats per ping-pong slot

typedef float v2f __attribute__((ext_vector_type(2)));
typedef float v8f __attribute__((ext_vector_type(8)));

static __device__ __forceinline__ v8f wmma4(v2f a, v2f b, v8f c) {
  // D(16x16,f32) = A(16x4,f32) * B(4x16,f32) + C
  return __builtin_amdgcn_wmma_f32_16x16x4_f32(false, a, false, b, (short)0, c, false, false);
}

static __device__ __forceinline__ float pick4(int j, float a0, float a1, float a2, float a3) {
  float r = (j == 0) ? a0 : a1;
  r = (j == 2) ? a2 : r;
  r = (j == 3) ? a3 : r;
  return r;
}

__global__ void __launch_bounds__(WGSZ, 1)
iva_iss_persistent(const float* __restrict__ X, float* __restrict__ Y,
                   float* __restrict__ wsf, unsigned* __restrict__ cnt)
{
  __shared__ float lds_r[WAVES_PER_WG][4];

  const int tid = threadIdx.x;
  const int l   = tid & 31;            // lane in wave32
  const int wv  = tid >> 5;
  const int p   = blockIdx.x * WAVES_PER_WG + wv;   // frequency pair id
  const int f1  = 2 * p;
  const int f2  = f1 + 1;
  const bool v1ok = (f1 < FF);
  const bool v2ok = (f2 < FF);

  const bool hi  = (l >= 16);
  const int  li  = l & 15;
  const int  myM = (l >> 2) & 3;       // source index of this lane's W element
  const int  myI = l & 3;              // column index of this lane's W element

  // per-lane W element: W[f = hi][myM][myI]; init W = I
  float Wre = (myM == myI) ? 1.f : 0.f;
  float Wim = 0.f;

  // U fragments: 16x16 = diag(R(U[m,f1]), R(U[m,f2])), C-layout; init identity
  v8f U0, U1, U2, U3;
  {
#pragma unroll
    for (int v = 0; v < 8; ++v) {
      float idv = (li == v + (hi ? 8 : 0)) ? 1.f : 0.f;
      U0[v] = idv; U1[v] = idv; U2[v] = idv; U3[v] = idv;
    }
  }

  const float2* X2 = (const float2*)X;
  float2* Y2 = (float2*)Y;

#pragma unroll 1
  for (int t = 0; t < TT; ++t) {
    // ---------------- load x_t for both freqs (wave-uniform) ----------------
    float x1r[4], x1i[4], x2r[4], x2i[4];
#pragma unroll
    for (int kk = 0; kk < KCH; ++kk) {
      float2 a = make_float2(0.f, 0.f), b = make_float2(0.f, 0.f);
      if (v1ok) a = X2[(size_t)(kk * TT + t) * FF + f1];
      if (v2ok) b = X2[(size_t)(kk * TT + t) * FF + f2];
      x1r[kk] = a.x; x1i[kk] = a.y; x2r[kk] = b.x; x2i[kk] = b.y;
      if (v1ok && (t + 1 < TT))
        __builtin_prefetch(&X2[(size_t)(kk * TT + t + 1) * FF + f1], 0, 0);
    }

    auto computeY = [&](float& yre, float& yim) {
      float ar = pick4(myI, x1r[0], x1r[1], x1r[2], x1r[3]);
      float ai = pick4(myI, x1i[0], x1i[1], x1i[2], x1i[3]);
      float br = pick4(myI, x2r[0], x2r[1], x2r[2], x2r[3]);
      float bi = pick4(myI, x2i[0], x2i[1], x2i[2], x2i[3]);
      float xr = hi ? br : ar;
      float xi = hi ? bi : ai;
      float pr = Wre * xr - Wim * xi;
      float pi = Wre * xi + Wim * xr;
      pr += __shfl_xor(pr, 1, 32); pi += __shfl_xor(pi, 1, 32);
      pr += __shfl_xor(pr, 2, 32); pi += __shfl_xor(pi, 2, 32);
      yre = pr; yim = pi;
    };

    // ---------------- y = W x; partial r over this wave's freqs ----------------
    float yre, yim;
    computeY(yre, yim);
    float p2 = yre * yre + yim * yim;
    p2 += __shfl_xor(p2, 16, 32);                 // sum both freqs of the pair
    if (!hi && myI == 0) lds_r[wv][myM] = p2;
    __syncthreads();

    const int slot = t & 1;
    float* pg = wsf + PART_BASE + slot * PART_STRIDE;
    if (tid < 4) {
      float s = 0.f;
#pragma unroll
      for (int w = 0; w < WAVES_PER_WG; ++w) s += lds_r[w][tid];
      pg[blockIdx.x * 4 + tid] = s;
    }
    __threadfence();
    __syncthreads();
    if (tid == 0) {
      __hip_atomic_fetch_add(cnt, 1u, __ATOMIC_ACQ_REL, __HIP_MEMORY_SCOPE_AGENT);
      const unsigned tgt = (unsigned)NWG * (unsigned)(t + 1);
      while (__hip_atomic_load(cnt, __ATOMIC_ACQUIRE, __HIP_MEMORY_SCOPE_AGENT) < tgt) {
        __builtin_amdgcn_s_sleep(2);
      }
    }
    __syncthreads();
    (void)__hip_atomic_load(cnt, __ATOMIC_ACQUIRE, __HIP_MEMORY_SCOPE_AGENT);

    // ---------------- final r, phi ----------------
    float rsum0 = 0.f, rsum1 = 0.f, rsum2 = 0.f, rsum3 = 0.f;
    const float4* p4 = (const float4*)pg;
#pragma unroll 1
    for (int w = 0; w < NWG; ++w) {
      float4 v = p4[w];
      rsum0 += v.x; rsum1 += v.y; rsum2 += v.z; rsum3 += v.w;
    }
    float sm[4];
    {
      float rs[4] = {rsum0, rsum1, rsum2, rsum3};
#pragma unroll
      for (int kk = 0; kk < 4; ++kk) {
        float r = sqrtf(fmaxf(rs[kk], EPSR));
        float phi = 1.f / (2.f * fmaxf(r, EPSR));
        sm[kk] = (1.f - ALPHA) * phi;
      }
    }

    // ---------------- U[m] = a*U[m] + (1-a)phi_m * R(x x^H)  (WMMA rank-4) ----------------
    v2f vb;
    {
      int jj = myI;
      bool lo4 = ((l & 4) == 0);
      float v1v = lo4 ? pick4(jj, x1r[0], x1r[1], x1r[2], x1r[3])
                      : pick4(jj, x1i[0], x1i[1], x1i[2], x1i[3]);
      float v2v = lo4 ? pick4(jj, x2r[0], x2r[1], x2r[2], x2r[3])
                      : pick4(jj, x2i[0], x2i[1], x2i[2], x2i[3]);
      float u1v = lo4 ? -pick4(jj, x1i[0], x1i[1], x1i[2], x1i[3])
                      :  pick4(jj, x1r[0], x1r[1], x1r[2], x1r[3]);
      float u2v = lo4 ? -pick4(jj, x2i[0], x2i[1], x2i[2], x2i[3])
                      :  pick4(jj, x2r[0], x2r[1], x2r[2], x2r[3]);
      bool inlo = (!hi) && (l < 8);
      bool inhi = hi && (l >= 24);
      vb[0] = inlo ? v1v : (inhi ? v2v : 0.f);
      vb[1] = inlo ? u1v : (inhi ? u2v : 0.f);
    }
#pragma unroll
    for (int m = 0; m < 4; ++m) {
      v8f& UF = (m == 0) ? U0 : (m == 1) ? U1 : (m == 2) ? U2 : U3;
      v2f a; a[0] = vb[0] * sm[m]; a[1] = vb[1] * sm[m];
      v8f c;
#pragma unroll
      for (int v = 0; v < 8; ++v) c[v] = UF[v] * ALPHA;
      UF = wmma4(a, vb, c);
    }

    // ---------------- ISS sweeps ----------------
#pragma unroll 1
    for (int it = 0; it < NIT; ++it) {
#pragma unroll 1
      for (int k = 0; k < KCH; ++k) {
        // broadcast w_k = row k of W (pre-update), both freqs
        float wr1[4], wi1[4], wr2[4], wi2[4];
#pragma unroll
        for (int i = 0; i < 4; ++i) {
          wr1[i] = __shfl(Wre, 4 * k + i, 32);
          wi1[i] = __shfl(Wim, 4 * k + i, 32);
          wr2[i] = __shfl(Wre, 16 + 4 * k + i, 32);
          wi2[i] = __shfl(Wim, 16 + 4 * k + i, 32);
        }
        float ws1[8], ws2[8];                    // R-vector of conj(w_k)
#pragma unroll
        for (int i = 0; i < 4; ++i) {
          ws1[i] = wr1[i]; ws1[4 + i] = -wi1[i];
          ws2[i] = wr2[i]; ws2[4 + i] = -wi2[i];
        }
        // B chunks (shared across m): cols 0-7 -> f1, cols 8-15 -> f2
        v2f Bc[4];
#pragma unroll
        for (int i = 0; i < 4; ++i) {
#pragma unroll
          for (int off = 0; off < 2; ++off) {
            const int idx0 = 4 * i + off;        // K-row when !hi
            const int idx1 = 4 * i + 2 + off;    // K-row when hi
            float vlo_f0 = (idx0 < 8)  ? ws1[idx0 & 7] : 0.f;
            float vhi_f0 = (idx1 < 8)  ? ws1[idx1 & 7] : 0.f;
            float vlo_f1 = (idx0 >= 8) ? ws2[idx0 & 7] : 0.f;
            float vhi_f1 = (idx1 >= 8) ? ws2[idx1 & 7] : 0.f;
            float vf0 = hi ? vhi_f0 : vlo_f0;
            float vf1 = hi ? vhi_f1 : vlo_f1;
            Bc[i][off] = (li >= 8) ? vf1 : vf0;
          }
        }

        float myvkre = 0.f, myvkim = 0.f;
#pragma unroll
        for (int m = 0; m < 4; ++m) {
          const v8f& UF = (m == 0) ? U0 : (m == 1) ? U1 : (m == 2) ? U2 : U3;
          float sw[8];
#pragma unroll
          for (int v = 0; v < 8; ++v) sw[v] = __shfl_xor(UF[v], 16, 32);
          // A chunks from symmetric C-layout fragment (R(U) symmetric since U Hermitian)
          v2f A0c, A1c, A2c, A3c;
          A0c[0] = hi ? sw[2] : UF[0]; A0c[1] = hi ? sw[3] : UF[1];
          A1c[0] = hi ? sw[6] : UF[4]; A1c[1] = hi ? sw[7] : UF[5];
          A2c[0] = hi ? UF[2] : sw[0]; A2c[1] = hi ? UF[3] : sw[1];
          A3c[0] = hi ? UF[6] : sw[4]; A3c[1] = hi ? UF[7] : sw[5];
          v8f acc = {0.f, 0.f, 0.f, 0.f, 0.f, 0.f, 0.f, 0.f};
          acc = wmma4(A0c, Bc[0], acc);
          acc = wmma4(A1c, Bc[1], acc);
          acc = wmma4(A2c, Bc[2], acc);
          acc = wmma4(A3c, Bc[3], acc);
          // temp(m, myF) in acc[0..7] on lanes 0-7 (f1) / 24-31 (f2)
          float den = 0.f, nre = 0.f, nim = 0.f;
#pragma unroll
          for (int i = 0; i < 4; ++i) {
            float wrl = hi ? wr2[i] : wr1[i];
            float wil = hi ? wi2[i] : wi1[i];
            float Wm1r = __shfl(Wre, 4 * m + i, 32);
            float Wm1i = __shfl(Wim, 4 * m + i, 32);
            float Wm2r = __shfl(Wre, 16 + 4 * m + i, 32);
            float Wm2i = __shfl(Wim, 16 + 4 * m + i, 32);
            float Wmr = hi ? Wm2r : Wm1r;
            float Wmi = hi ? Wm2i : Wm1i;
            float tr = acc[i], ti = acc[4 + i];
            den += wrl * tr - wil * ti;
            nre += Wmr * tr - Wmi * ti;
            nim += Wmr * ti + Wmi * tr;
          }
          den = fmaxf(den, EPSR);
          float vkre = nre / den, vkim = nim / den;
          float alt = 1.f - (1.f / sqrtf(den));
          bool isk = (m == k);
          vkre = isk ? alt : vkre;
          vkim = isk ? 0.f : vkim;
          int srcl = hi ? 24 : 0;
          float bre = __shfl(vkre, srcl, 32);
          float bim = __shfl(vkim, srcl, 32);
          if (m == myM) { myvkre = bre; myvkim = bim; }
        }
        // W[f,m,i] -= vk[m,f] * w_k[f,i]
        int wsrc = (l & 16) + 4 * k + myI;
        float wkr = __shfl(Wre, wsrc, 32);
        float wki = __shfl(Wim, wsrc, 32);
        Wre -= myvkre * wkr - myvkim * wki;
        Wim -= myvkre * wki + myvkim * wkr;
      }
    }

    // ---------------- y_out = W x ; store ----------------
    computeY(yre, yim);
    int f = f1 + (hi ? 1 : 0);
    if (myI == 0 && f < FF) {
      Y2[(size_t)(myM * TT + t) * FF + f] = make_float2(yre, yim);
    }
  }
}

__global__ void iva_init(float* wsf) {
  int i = blockIdx.x * blockDim.x + threadIdx.x;
  if (i < 1024) wsf[i] = 0.f;
}

extern "C" void kernel_launch(void* const* d_in, const int* in_sizes, int n_in,
                              void* d_out, int out_size, void* d_ws, size_t ws_size,
                              hipStream_t stream) {
  const float* X = (const float*)d_in[0];
  float* Y = (float*)d_out;
  float* wsf = (float*)d_ws;
  unsigned* cnt = (unsigned*)d_ws;
  hipLaunchKernelGGL(iva_init, dim3(4), dim3(256), 0, stream, wsf);
  hipLaunchKernelGGL(iva_iss_persistent, dim3(NWG), dim3(WGSZ), 0, stream, X, Y, wsf, cnt);
}